// GCPNet_14431090114651
// MI455X (gfx1250) — compile-verified
//
#include <hip/hip_runtime.h>

#define E   128
#define NV  8192
#define CNC 512
#define GG  64
#define TMAX 32

typedef __attribute__((ext_vector_type(2))) float v2f;
typedef __attribute__((ext_vector_type(8))) float v8f;

// ---------------------------------------------------------------------------
// Generic f32 WMMA GEMM:  D[m,n] = act( sum_k A[m,k] * Bmat[k,n] + extras )
//   Bmat[k,n] = BTRANS ? B[n*ldb + k] : B[k*ldb + n]
// One wave computes one 16x16 output tile using V_WMMA_F32_16X16X4_F32.
// Block = 256 threads (8 waves); grid = (M/16, (N/16)/8).
// ---------------------------------------------------------------------------
template <bool BTRANS, bool RELU, bool HASBIAS, bool RANK1>
__global__ __launch_bounds__(256) void gemm_f32_wmma(
    const float* __restrict__ A, int lda,
    const float* __restrict__ B, int ldb,
    const float* __restrict__ bias,
    const float* __restrict__ rvec, const float* __restrict__ cvec,
    float* __restrict__ D, int ldd, int K)
{
    const int lane = threadIdx.x & 31;
    const int wave = threadIdx.x >> 5;
    const int m0 = blockIdx.x * 16;
    const int n0 = (blockIdx.y * 8 + wave) * 16;
    const int ln = lane & 15;
    const int hi = lane >> 4;          // 0 for lanes 0-15, 1 for lanes 16-31
    const int koff = hi * 2;

    v8f acc;
#pragma unroll
    for (int v = 0; v < 8; ++v) acc[v] = 0.0f;

    const int am = m0 + ln;
    const int bn = n0 + ln;

    for (int k0 = 0; k0 < K; k0 += 4) {
        const int ak = k0 + koff;
        v2f a, b;
        a.x = A[(size_t)am * lda + ak];
        a.y = A[(size_t)am * lda + ak + 1];
        if (BTRANS) {
            b.x = B[(size_t)bn * ldb + ak];
            b.y = B[(size_t)bn * ldb + ak + 1];
        } else {
            b.x = B[(size_t)ak * ldb + bn];
            b.y = B[(size_t)(ak + 1) * ldb + bn];
        }
        acc = __builtin_amdgcn_wmma_f32_16x16x4_f32(
            false, a, false, b, (short)0, acc, false, false);
    }

#pragma unroll
    for (int v = 0; v < 8; ++v) {
        const int row = m0 + v + hi * 8;
        const int col = n0 + ln;
        float x = acc[v];
        if (HASBIAS) x += bias[col];
        if (RANK1)   x += rvec[row] * cvec[col];
        if (RELU)    x = fmaxf(x, 0.0f);
        D[(size_t)row * ldd + col] = x;
    }
}

// ---------------------------------------------------------------------------
// Fused LSTM step:  gates = xb + hin @ Whh.T ; (hout, cv) = lstm(gates, cv).
// One block = 16-row stripe of hv, all 512 gate columns.
// Wave w computes tiles at columns {w*16, w*16+128, w*16+256, w*16+384}
// => per lane, acc[0..3][v] are exactly (i,f,g,o) for row m0+v+hi*8,
//    col j = w*16+ln -> gate update fully in registers, no gbuf round trip.
// A stripe staged in LDS (padded stride 132 to avoid bank conflicts),
// reused by 4 WMMAs per K-step (register blocking on N).
// ---------------------------------------------------------------------------
#define AS_LD 132
__global__ __launch_bounds__(256) void lstm_step_fused(
    const float* __restrict__ hin,   // [NV, E]
    const float* __restrict__ Whh,   // [4E, E] row-major
    const float* __restrict__ xb,    // [NV, 4E], bias pre-added
    float* __restrict__ cv,          // [NV, E] in/out
    float* __restrict__ hout)        // [NV, E]
{
    __shared__ float As[16 * AS_LD];
    const int tid  = threadIdx.x;
    const int lane = tid & 31;
    const int wave = tid >> 5;
    const int m0 = blockIdx.x * 16;
    const int ln = lane & 15;
    const int hi = lane >> 4;
    const int koff = hi * 2;
    const int j0 = wave * 16;        // base column within one gate block

    // Stage the 16x128 A stripe (2048 contiguous floats) into padded LDS.
    const float* src = hin + (size_t)m0 * E;
#pragma unroll
    for (int t = 0; t < 8; ++t) {
        const int e = tid + t * 256;
        As[(e >> 7) * AS_LD + (e & 127)] = src[e];
    }

    // Prefetch the cv line needed only in the epilogue (global_prefetch_b8).
    __builtin_prefetch(&cv[(size_t)(m0 + ln) * E + j0], 1, 3);

    // Init accumulators from the loop-invariant xb (i,f,g,o tiles).
    v8f acc[4];
#pragma unroll
    for (int g = 0; g < 4; ++g)
#pragma unroll
        for (int v = 0; v < 8; ++v) {
            const int row = m0 + v + hi * 8;
            acc[g][v] = xb[(size_t)row * (4 * E) + j0 + g * E + ln];
        }

    __syncthreads();

    for (int k0 = 0; k0 < E; k0 += 4) {
        const int ak = k0 + koff;
        v2f a;
        a.x = As[ln * AS_LD + ak];
        a.y = As[ln * AS_LD + ak + 1];
#pragma unroll
        for (int g = 0; g < 4; ++g) {
            const int brow = j0 + g * E + ln;   // Whh row == gate output col
            v2f b;
            b.x = Whh[(size_t)brow * E + ak];
            b.y = Whh[(size_t)brow * E + ak + 1];
            acc[g] = __builtin_amdgcn_wmma_f32_16x16x4_f32(
                false, a, false, b, (short)0, acc[g], false, false);
        }
    }

    // Fused gate nonlinearities + state update, all in registers.
#pragma unroll
    for (int v = 0; v < 8; ++v) {
        const int row = m0 + v + hi * 8;
        const int j = j0 + ln;
        const float i_ = 1.0f / (1.0f + __expf(-acc[0][v]));
        const float f_ = 1.0f / (1.0f + __expf(-acc[1][v]));
        const float g_ = tanhf(acc[2][v]);
        const float o_ = 1.0f / (1.0f + __expf(-acc[3][v]));
        const size_t idx = (size_t)row * E + j;
        const float c2 = f_ * cv[idx] + i_ * g_;
        cv[idx] = c2;
        hout[idx] = o_ * tanhf(c2);
    }
}

// ---------------------------------------------------------------------------
// Row-sum of M_vv (8192 x 8192): one wave per row, float4 loads, shfl reduce.
// ---------------------------------------------------------------------------
__global__ __launch_bounds__(256) void rowsum_kernel(
    const float* __restrict__ M, float* __restrict__ r)
{
    const int row  = blockIdx.x * 8 + (threadIdx.x >> 5);
    const int lane = threadIdx.x & 31;
    const float4* p = (const float4*)(M + (size_t)row * NV);
    float s = 0.0f;
    for (int t = lane; t < NV / 4; t += 32) {
        float4 v = p[t];
        s += v.x + v.y + v.z + v.w;
    }
#pragma unroll
    for (int off = 16; off > 0; off >>= 1) s += __shfl_xor(s, off, 32);
    if (lane == 0) r[row] = s;
}

__global__ void compute_v0(const float* __restrict__ w,
                           const float* __restrict__ b,
                           float* __restrict__ v0)
{
    const int j = threadIdx.x;
    v0[j] = w[j] + b[j];
}

// bias_v[j] = bih[j] + bhh[j];  w0[j] = sum_k Wih_v[j, k<E] * v0[k]
__global__ void compute_bias_w0(const float* __restrict__ Wih,
                                const float* __restrict__ bih,
                                const float* __restrict__ bhh,
                                const float* __restrict__ v0,
                                float* __restrict__ bias_v,
                                float* __restrict__ w0)
{
    const int j = blockIdx.x * blockDim.x + threadIdx.x;
    if (j >= 4 * E) return;
    bias_v[j] = bih[j] + bhh[j];
    float s = 0.0f;
    for (int k = 0; k < E; ++k) s += Wih[(size_t)j * (2 * E) + k] * v0[k];
    w0[j] = s;
}

__global__ void init_h_and_outV(const float* __restrict__ v0,
                                float* __restrict__ h,
                                float* __restrict__ c,
                                float* __restrict__ outV)
{
    const int idx = blockIdx.x * blockDim.x + threadIdx.x;
    if (idx >= NV * E) return;
    const float v = v0[idx & (E - 1)];
    h[idx] = v;
    c[idx] = 0.0f;
    outV[idx] = v;
}

__global__ __launch_bounds__(128) void vote_final(
    const float* __restrict__ h2, const float* __restrict__ W3,
    const float* __restrict__ b3, const int* __restrict__ slice_idx,
    float* __restrict__ out)
{
    __shared__ float red[128];
    const int g = blockIdx.x;
    const int v = threadIdx.x;
    const float* hr = h2 + ((size_t)g * 128 + v) * E;
    float s = 0.0f;
    for (int k = 0; k < E; ++k) s += hr[k] * W3[k];
    s += b3[0];
    red[v] = s;
    __syncthreads();
    for (int off = 64; off > 0; off >>= 1) {
        if (v < off) red[v] += red[v + off];
        __syncthreads();
    }
    if (v == 0) {
        const float mean = red[0] / (float)slice_idx[g];
        out[g] = 1.0f / (1.0f + __expf(-mean));
    }
}

// ---------------------------------------------------------------------------
extern "C" void kernel_launch(void* const* d_in, const int* in_sizes, int n_in,
                              void* d_out, int out_size, void* d_ws, size_t ws_size,
                              hipStream_t stream)
{
    (void)in_sizes; (void)n_in; (void)out_size; (void)ws_size;

    const float* M_vv    = (const float*)d_in[0];
    const float* M_vc    = (const float*)d_in[1];
    const float* C       = (const float*)d_in[2];
    const int*   sliceIx = (const int*)  d_in[3];
    const float* v_emb_w = (const float*)d_in[4];
    const float* v_emb_b = (const float*)d_in[5];
    const float* mlpC_W  = (const float*)d_in[6];   // [3,E,E]
    const float* mlpC_b  = (const float*)d_in[7];   // [3,E]
    const float* Wih_v   = (const float*)d_in[10];  // [4E, 2E]
    const float* Whh_v   = (const float*)d_in[11];  // [4E, E]
    const float* bih_v   = (const float*)d_in[12];
    const float* bhh_v   = (const float*)d_in[13];
    const float* vote_W1 = (const float*)d_in[18];
    const float* vote_b1 = (const float*)d_in[19];
    const float* vote_W2 = (const float*)d_in[20];
    const float* vote_b2 = (const float*)d_in[21];
    const float* vote_W3 = (const float*)d_in[22];  // [1,E]
    const float* vote_b3 = (const float*)d_in[23];

    float* out = (float*)d_out;           // [64] ++ [NV*E] ++ [CNC*E]

    // -------- workspace layout (floats) --------
    float* ws = (float*)d_ws;
    float* r      = ws;                    ws += NV;
    float* v0     = ws;                    ws += 128;
    float* bias_v = ws;                    ws += 512;
    float* w0     = ws;                    ws += 512;
    float* H1     = ws;                    ws += CNC * E;
    float* H2     = ws;                    ws += CNC * E;
    float* mlpC   = ws;                    ws += CNC * E;
    float* P      = ws;                    ws += NV * E;      // reused as vote h1
    float* xb     = ws;                    ws += NV * 4 * E;  // loop-invariant gates
    float* hva    = ws;                    ws += NV * E;      // hv double buffer A
    float* hvb    = ws;                    ws += NV * E;      // hv double buffer B
    float* cv     = ws;                    ws += NV * E;
    float* h2v    = ws;                    ws += NV * E;      // vote h2

    // 1) rowsum(M_vv) -> r  (rank-1 collapse of M_vv @ V)
    rowsum_kernel<<<NV / 8, 256, 0, stream>>>(M_vv, r);

    // 2) v0, bias_v, w0 = Wih_v[:, :E] @ v0
    compute_v0<<<1, 128, 0, stream>>>(v_emb_w, v_emb_b, v0);
    compute_bias_w0<<<2, 256, 0, stream>>>(Wih_v, bih_v, bhh_v, v0, bias_v, w0);

    // 3) mlp3(C): three [512,128]x[128,128] WMMA GEMMs
    gemm_f32_wmma<true, true, true, false><<<dim3(CNC / 16, 1), 256, 0, stream>>>(
        C, E, mlpC_W + 0 * E * E, E, mlpC_b + 0 * E, nullptr, nullptr, H1, E, E);
    gemm_f32_wmma<true, true, true, false><<<dim3(CNC / 16, 1), 256, 0, stream>>>(
        H1, E, mlpC_W + 1 * E * E, E, mlpC_b + 1 * E, nullptr, nullptr, H2, E, E);
    gemm_f32_wmma<true, false, true, false><<<dim3(CNC / 16, 1), 256, 0, stream>>>(
        H2, E, mlpC_W + 2 * E * E, E, mlpC_b + 2 * E, nullptr, nullptr, mlpC, E, E);

    // 4) P = M_vc @ mlpC   ([8192,512] x [512,128])
    gemm_f32_wmma<false, false, false, false><<<dim3(NV / 16, 1), 256, 0, stream>>>(
        M_vc, CNC, mlpC, E, nullptr, nullptr, nullptr, P, E, CNC);

    // 5) xb = P @ Wih_v[:, E:2E].T + (bih+bhh) + r (x) w0   -> [8192, 512]
    gemm_f32_wmma<true, false, true, true><<<dim3(NV / 16, 4), 256, 0, stream>>>(
        P, E, Wih_v + E, 2 * E, bias_v, r, w0, xb, 4 * E, E);

    // 6) hv = broadcast(v0), cv = 0, and write V output block
    init_h_and_outV<<<(NV * E) / 256, 256, 0, stream>>>(v0, hva, cv, out + GG);

    // 7) 32 fused LSTM steps with double-buffered hv (result ends in hva)
    float* bufs[2] = { hva, hvb };
    for (int t = 0; t < TMAX; ++t) {
        lstm_step_fused<<<NV / 16, 256, 0, stream>>>(
            bufs[t & 1], Whh_v, xb, cv, bufs[(t + 1) & 1]);
    }

    // 8) vote MLP: reuse P as h1
    gemm_f32_wmma<true, true, true, false><<<dim3(NV / 16, 1), 256, 0, stream>>>(
        hva, E, vote_W1, E, vote_b1, nullptr, nullptr, P, E, E);
    gemm_f32_wmma<true, true, true, false><<<dim3(NV / 16, 1), 256, 0, stream>>>(
        P, E, vote_W2, E, vote_b2, nullptr, nullptr, h2v, E, E);
    vote_final<<<GG, 128, 0, stream>>>(h2v, vote_W3, vote_b3, sliceIx, out);

    // 9) C passthrough output block
    hipMemcpyAsync(out + GG + (size_t)NV * E, (void*)C,
                   (size_t)CNC * E * sizeof(float),
                   hipMemcpyDeviceToDevice, stream);
}